// WeightingRouter_43739946942519
// MI455X (gfx1250) — compile-verified
//
#include <hip/hip_runtime.h>

// Elementwise "weighting router":
//   h = x[:,0]
//   h >= 0.3 : [ (h-0.3)/0.7, 0,           (1-h)/0.7 ]
//   h <  0.3 : [ 0,           (0.3-h)/0.3,  h/0.3    ]
//
// Memory-bound: 20 B/row * 16.7M rows ~= 335 MB, floor ~14.4 us at 23.3 TB/s.
// Strategy: 4 rows/thread -> 2x B128 NT loads + 3x B128 NT stores, branch-free
// v_cndmask math, reciprocal-multiply instead of division.

typedef float v4f __attribute__((ext_vector_type(4)));

static constexpr float kThr   = 0.3f;
static constexpr float kInvHi = 1.0f / (1.0f - 0.3f);  // 1/0.7
static constexpr float kInvLo = 1.0f / 0.3f;           // 1/0.3

__global__ __launch_bounds__(256) void router_quad(const v4f* __restrict__ x,
                                                   v4f* __restrict__ out,
                                                   int nquads) {
  int t = blockIdx.x * blockDim.x + threadIdx.x;
  if (t >= nquads) return;

  long xi = 2L * t;          // two float4 = 4 rows of (h, w)
  v4f a = __builtin_nontemporal_load(x + xi);
  v4f b = __builtin_nontemporal_load(x + xi + 1);

  float h[4] = {a.x, a.z, b.x, b.z};
  float r[12];
#pragma unroll
  for (int i = 0; i < 4; ++i) {
    float hv = h[i];
    bool hi = hv >= kThr;
    r[3 * i + 0] = hi ? (hv - kThr) * kInvHi : 0.0f;
    r[3 * i + 1] = hi ? 0.0f : (kThr - hv) * kInvLo;
    r[3 * i + 2] = hi ? (1.0f - hv) * kInvHi : hv * kInvLo;
  }

  v4f s0 = {r[0], r[1], r[2],  r[3]};
  v4f s1 = {r[4], r[5], r[6],  r[7]};
  v4f s2 = {r[8], r[9], r[10], r[11]};

  long o = 3L * t;           // 12 contiguous output floats per thread
  __builtin_nontemporal_store(s0, out + o);
  __builtin_nontemporal_store(s1, out + o + 1);
  __builtin_nontemporal_store(s2, out + o + 2);
}

// Scalar tail for rows % 4 (empty for the reference's N = 16,777,216).
__global__ void router_tail(const float* __restrict__ x, float* __restrict__ out,
                            long row0, long rows) {
  long r = row0 + (long)threadIdx.x;
  if (r >= rows) return;
  float hv = x[2 * r];
  bool hi = hv >= kThr;
  out[3 * r + 0] = hi ? (hv - kThr) * kInvHi : 0.0f;
  out[3 * r + 1] = hi ? 0.0f : (kThr - hv) * kInvLo;
  out[3 * r + 2] = hi ? (1.0f - hv) * kInvHi : hv * kInvLo;
}

extern "C" void kernel_launch(void* const* d_in, const int* in_sizes, int n_in,
                              void* d_out, int out_size, void* d_ws, size_t ws_size,
                              hipStream_t stream) {
  const float* x = (const float*)d_in[0];
  float* out = (float*)d_out;

  long rows   = (long)in_sizes[0] / 2;  // in_sizes[0] = N*2 flat floats
  long nquads = rows / 4;
  long tail   = rows - nquads * 4;

  if (nquads > 0) {
    const int threads = 256;  // 8 wave32 waves per block
    unsigned blocks = (unsigned)((nquads + threads - 1) / threads);
    router_quad<<<blocks, threads, 0, stream>>>(
        (const v4f*)x, (v4f*)out, (int)nquads);
  }
  if (tail > 0) {
    router_tail<<<1, (int)tail, 0, stream>>>(x, out, nquads * 4, rows);
  }
}